// ConvertParamsTEtoParams0TE_16277926052109
// MI455X (gfx1250) — compile-verified
//
#include <hip/hip_runtime.h>
#include <hip/hip_bf16.h>
#include <math.h>

// Problem sizes (fixed by the reference)
#define BB 4096
#define NH 64
#define NV 256
#define LDS_STRIDE 68   // 64 + 4 pad: A-column reads hit distinct banks (4*row+k mod 64)

typedef __attribute__((ext_vector_type(2))) float v2f;
typedef __attribute__((ext_vector_type(8))) float v8f;

__global__ __launch_bounds__(256) void convert_params_kernel(
    const float* __restrict__ bTE1,       // [B, NV]
    const float* __restrict__ wtTE1,      // [B, NH, NV]
    const float* __restrict__ muh1,       // [B, NH]
    const float* __restrict__ wt1,        // [B, NH, NV]
    const float* __restrict__ muhTE1,     // [B, NH]
    const float* __restrict__ covh_diag1, // [B, NH]
    const float* __restrict__ covhTE1,    // [B, NH, NH]
    const float* __restrict__ sig2TE,     // [B]
    float* __restrict__ out_sig,          // [B]
    float* __restrict__ out_bTE2,         // [B, NV]
    float* __restrict__ out_wtTE2)        // [B, NH, NV]
{
    const int b   = blockIdx.x;
    const int tid = threadIdx.x;            // 0..255 (8 waves of 32)

    __shared__ float lds_covh[NH * LDS_STRIDE]; // 64x64 A tile, padded
    __shared__ float lds_s[NH];                 // sqrt(diag)
    __shared__ float lds_hinv[NH];              // 0.5 / sqrt(diag)
    __shared__ float lds_muh[NH];
    __shared__ float lds_muhTE[NH];

    const size_t bHV = (size_t)b * NH * NV;
    const float* __restrict__ covh_b = covhTE1 + (size_t)b * NH * NH;
    const float* __restrict__ wt_b   = wt1   + bHV;
    const float* __restrict__ wtTE_b = wtTE1 + bHV;
    float* __restrict__ outw_b       = out_wtTE2 + bHV;

    // ---- Stage covhTE1[b] (16 KB) into LDS, 16 elements per thread ----
    #pragma unroll
    for (int i = tid; i < NH * NH; i += 256) {
        const int r = i >> 6;
        const int c = i & 63;
        lds_covh[r * LDS_STRIDE + c] = covh_b[i];
    }
    if (tid < NH) {
        const float d  = covh_diag1[(size_t)b * NH + tid];
        const float sv = sqrtf(d);
        lds_s[tid]     = sv;
        lds_hinv[tid]  = 0.5f / sv;
        lds_muh[tid]   = muh1[(size_t)b * NH + tid];
        lds_muhTE[tid] = muhTE1[(size_t)b * NH + tid];
    }
    if (tid == 0) out_sig[b] = sig2TE[b];
    __syncthreads();

    // ---- bTE2: one output column per thread (coalesced over tid) ----
    {
        float acc = bTE1[(size_t)b * NV + tid];
        #pragma unroll 4
        for (int h = 0; h < NH; ++h) {
            acc += wtTE_b[h * NV + tid] * lds_muh[h];
            acc += wt_b[h * NV + tid]   * lds_muhTE[h];
        }
        out_bTE2[(size_t)b * NV + tid] = acc;
    }

    // ---- wtTE2 via V_WMMA_F32_16X16X4_F32 ----
    // 64x256 output = 4 m-tiles x 16 n-tiles of 16x16. 8 waves, 8 tiles each.
    const int wave  = tid >> 5;
    const int lane  = tid & 31;
    const int lhalf = lane >> 4;   // 0: lanes 0-15, 1: lanes 16-31
    const int l16   = lane & 15;
    const int m0    = (wave & 3) * 16;       // wave's m-tile origin
    const int nc0   = (wave >> 2) * 8;       // first of 8 n-tiles

    const float* aRow = &lds_covh[(m0 + l16) * LDS_STRIDE]; // this lane's A row

    for (int tc = 0; tc < 8; ++tc) {
        const int n0 = (nc0 + tc) * 16;
        v8f acc = {};
        #pragma unroll
        for (int kk = 0; kk < NH; kk += 4) {
            const int ka = kk + 2 * lhalf;          // ISA: lane-half selects K / K+2
            v2f afrag, bfrag;
            afrag.x = aRow[ka];                     // A: 16x4, M=lane, VGPR = K parity
            afrag.y = aRow[ka + 1];
            bfrag.x = wt_b[(size_t)ka * NV + n0 + l16];       // B: 4x16, N=lane
            bfrag.y = wt_b[(size_t)(ka + 1) * NV + n0 + l16];
            acc = __builtin_amdgcn_wmma_f32_16x16x4_f32(
                /*neg_a=*/false, afrag, /*neg_b=*/false, bfrag,
                /*c_mod=*/(short)0, acc, /*reuse_a=*/false, /*reuse_b=*/false);
        }
        // C/D layout: VGPR r -> M = m0 + r + 8*lhalf, N = n0 + l16
        #pragma unroll
        for (int r = 0; r < 8; ++r) {
            const int m = m0 + r + 8 * lhalf;
            const float sv = lds_s[m];
            const float hv = lds_hinv[m];
            const size_t idx = (size_t)m * NV + n0 + l16;
            outw_b[idx] = acc[r] * hv + sv * wtTE_b[idx];
        }
    }
}

extern "C" void kernel_launch(void* const* d_in, const int* in_sizes, int n_in,
                              void* d_out, int out_size, void* d_ws, size_t ws_size,
                              hipStream_t stream) {
    const float* bTE1       = (const float*)d_in[0];
    const float* wtTE1      = (const float*)d_in[1];
    const float* muh1       = (const float*)d_in[2];
    const float* wt1        = (const float*)d_in[3];
    const float* muhTE1     = (const float*)d_in[4];
    const float* covh_diag1 = (const float*)d_in[5];
    const float* covhTE1    = (const float*)d_in[6];
    const float* sig2TE     = (const float*)d_in[7];

    float* out       = (float*)d_out;
    float* out_sig   = out;                          // [B]
    float* out_bTE2  = out + BB;                     // [B, NV]
    float* out_wtTE2 = out + BB + (size_t)BB * NV;   // [B, NH, NV]

    convert_params_kernel<<<dim3(BB), dim3(256), 0, stream>>>(
        bTE1, wtTE1, muh1, wt1, muhTE1, covh_diag1, covhTE1, sig2TE,
        out_sig, out_bTE2, out_wtTE2);
}